// CustomLoss_22454089023928
// MI455X (gfx1250) — compile-verified
//
#include <hip/hip_runtime.h>

// MI455X / gfx1250: wave32, WMMA 16x16x4 f32 path for the separable Gaussian.
typedef __attribute__((ext_vector_type(2))) float v2f;
typedef __attribute__((ext_vector_type(8))) float v8f;

#define IMG   512
#define NCH   64
#define TILE  32
#define RAD   5
#define INR   42          // TILE + 2*RAD
#define INC   48          // padded to 3*16 columns
#define MIDR  32
#define MIDC  48
#define NPIX  (NCH * IMG * IMG)   // 16777216
#define C1F   0.0004f     // (0.01*2)^2
#define C2F   0.0036f     // (0.03*2)^2

// 11-tap Gaussian, sigma=1.5, truncate=3.5 (matches skimage/ref, normalized)
__constant__ float cK1D[11] = {
    0.0010284f, 0.0075988f, 0.0360008f, 0.1093607f, 0.2130055f,
    0.2660117f,
    0.2130055f, 0.1093607f, 0.0360008f, 0.0075988f, 0.0010284f
};

__device__ __forceinline__ v8f wmma4(v2f a, v2f b, v8f c) {
    // D = A(16x4) * B(4x16) + C(16x16), f32, wave32
    return __builtin_amdgcn_wmma_f32_16x16x4_f32(
        /*neg_a=*/false, a, /*neg_b=*/false, b,
        /*c_mod=*/(short)0, c, /*reuse_a=*/false, /*reuse_b=*/false);
}

extern "C" __global__ void __launch_bounds__(256)
ssim_tile_kernel(const float* __restrict__ pred,
                 const float* __restrict__ targ,
                 float* __restrict__ partial) {
    __shared__ float sp[INR * INC];          // pred tile + halo
    __shared__ float st[INR * INC];          // target tile + halo
    __shared__ float smid[5 * MIDR * MIDC];  // 5 vertically filtered fields
    __shared__ float kwt[64];                // zero-padded weight band table
    __shared__ float redA[8], redB[8];

    const int tid = threadIdx.x;
    const int blk = blockIdx.x;
    const int ch  = blk >> 8;                 // 256 tiles per channel
    const int t   = blk & 255;
    const int r0  = (t >> 4) * TILE;
    const int c0  = (t & 15) * TILE;
    const float* pc = pred + (size_t)ch * (IMG * IMG);
    const float* tc = targ + (size_t)ch * (IMG * IMG);

    // weight table: kwt[16+d] = k1d[d] for d in [0,10], else 0
    if (tid < 64) {
        float w = 0.0f;
        if (tid >= 16 && tid <= 26) w = cK1D[tid - 16];
        kwt[tid] = w;
    }

    // stage halo tile with edge clamp (matches 'edge' padding)
    for (int i = tid; i < INR * INC; i += 256) {
        int rr = i / INC, cc = i - rr * INC;
        int gr = r0 - RAD + rr; gr = gr < 0 ? 0 : (gr > IMG - 1 ? IMG - 1 : gr);
        int gc = c0 - RAD + cc; gc = gc < 0 ? 0 : (gc > IMG - 1 ? IMG - 1 : gc);
        sp[i] = pc[gr * IMG + gc];
        st[i] = tc[gr * IMG + gc];
    }
    __syncthreads();

    // ---- MSE over this block's 32x32 interior (each pixel owned once) ----
    float mse = 0.0f;
    for (int i = tid; i < TILE * TILE; i += 256) {
        int r = i >> 5, c = i & 31;
        float d = sp[(r + RAD) * INC + (c + RAD)] - st[(r + RAD) * INC + (c + RAD)];
        mse += d * d;
    }

    const int wave = tid >> 5;
    const int lane = tid & 31;
    const int m  = lane & 15;
    const int hi = lane >> 4;
    const v8f vzero = {0.f, 0.f, 0.f, 0.f, 0.f, 0.f, 0.f, 0.f};

    // ---- vertical pass: mid(16x16) = Band(16x28) @ in(28x16), 7 WMMAs ----
    if (wave < 6) {                     // 2 row-tiles x 3 col-tiles, one per wave
        const int rt = wave / 3;
        const int ct = wave - rt * 3;
        v8f acc[5];
        #pragma unroll
        for (int q = 0; q < 5; ++q) acc[q] = vzero;

        #pragma unroll
        for (int k0 = 0; k0 < 28; k0 += 4) {
            const int j0 = k0 + 2 * hi;
            v2f a;                      // banded weights: Wv[m][j] = k1d[j-m]
            a.x = kwt[16 + j0 - m];
            a.y = kwt[17 + j0 - m];
            int rr0 = rt * 16 + j0;     if (rr0 > INR - 1) rr0 = INR - 1;
            int rr1 = rt * 16 + j0 + 1; if (rr1 > INR - 1) rr1 = INR - 1;
            const int cc = ct * 16 + m;
            const float p0 = sp[rr0 * INC + cc], p1 = sp[rr1 * INC + cc];
            const float t0 = st[rr0 * INC + cc], t1 = st[rr1 * INC + cc];
            v2f b;
            b.x = p0;      b.y = p1;      acc[0] = wmma4(a, b, acc[0]);   // x
            b.x = t0;      b.y = t1;      acc[1] = wmma4(a, b, acc[1]);   // y
            b.x = p0 * p0; b.y = p1 * p1; acc[2] = wmma4(a, b, acc[2]);   // xx
            b.x = t0 * t0; b.y = t1 * t1; acc[3] = wmma4(a, b, acc[3]);   // yy
            b.x = p0 * t0; b.y = p1 * t1; acc[4] = wmma4(a, b, acc[4]);   // xy
        }
        #pragma unroll
        for (int q = 0; q < 5; ++q) {
            #pragma unroll
            for (int v = 0; v < 8; ++v) {
                const int mr = rt * 16 + v + 8 * hi;   // C/D layout
                const int mc = ct * 16 + m;
                smid[q * (MIDR * MIDC) + mr * MIDC + mc] = acc[q][v];
            }
        }
    }
    __syncthreads();

    // ---- horizontal pass + fused SSIM: out(16x16) = mid(16x28) @ Band(28x16) ----
    float ssum = 0.0f;
    if (wave < 4) {                     // 2x2 subtiles of the 32x32 output
        const int rt = wave >> 1;
        const int ct = wave & 1;
        v8f acc[5];
        #pragma unroll
        for (int q = 0; q < 5; ++q) acc[q] = vzero;

        #pragma unroll
        for (int k0 = 0; k0 < 28; k0 += 4) {
            const int j0 = k0 + 2 * hi;
            v2f b;                      // Wh[j][n] = k1d[j-n]
            b.x = kwt[16 + j0 - m];
            b.y = kwt[17 + j0 - m];
            const int row = rt * 16 + m;
            const int col = ct * 16 + j0;        // <= 43 < 48, in range
            #pragma unroll
            for (int q = 0; q < 5; ++q) {
                const int base = q * (MIDR * MIDC) + row * MIDC + col;
                v2f a;
                a.x = smid[base];
                a.y = smid[base + 1];
                acc[q] = wmma4(a, b, acc[q]);
            }
        }
        #pragma unroll
        for (int v = 0; v < 8; ++v) {
            const float ux  = acc[0][v], uy  = acc[1][v];
            const float uxx = acc[2][v], uyy = acc[3][v], uxy = acc[4][v];
            const float vx  = uxx - ux * ux;
            const float vy  = uyy - uy * uy;
            const float vxy = uxy - ux * uy;
            const float num = (2.0f * ux * uy + C1F) * (2.0f * vxy + C2F);
            const float den = (ux * ux + uy * uy + C1F) * (vx + vy + C2F);
            ssum += num / den;
        }
    }

    // ---- deterministic in-block reduction (fixed tree order) ----
    #pragma unroll
    for (int off = 16; off > 0; off >>= 1) {
        mse  += __shfl_down(mse,  (unsigned)off, 32);
        ssum += __shfl_down(ssum, (unsigned)off, 32);
    }
    if (lane == 0) { redA[wave] = mse; redB[wave] = ssum; }
    __syncthreads();
    if (tid == 0) {
        float sm = 0.0f, ss = 0.0f;
        #pragma unroll
        for (int w = 0; w < 8; ++w) { sm += redA[w]; ss += redB[w]; }
        partial[2 * blk]     = sm;
        partial[2 * blk + 1] = ss;
    }
}

extern "C" __global__ void __launch_bounds__(256)
ssim_reduce_kernel(const float* __restrict__ partial, int nblk,
                   float* __restrict__ out) {
    __shared__ double rm[256], rs[256];
    const int tid = threadIdx.x;
    double sm = 0.0, ss = 0.0;
    for (int i = tid; i < nblk; i += 256) {
        sm += (double)partial[2 * i];
        ss += (double)partial[2 * i + 1];
    }
    rm[tid] = sm; rs[tid] = ss;
    __syncthreads();
    if (tid == 0) {
        double a = 0.0, b = 0.0;
        for (int i = 0; i < 256; ++i) { a += rm[i]; b += rs[i]; }
        const double inv = 1.0 / (double)NPIX;
        out[0] = (float)(a * inv + 1.0 - b * inv);
    }
}

extern "C" void kernel_launch(void* const* d_in, const int* in_sizes, int n_in,
                              void* d_out, int out_size, void* d_ws, size_t ws_size,
                              hipStream_t stream) {
    (void)in_sizes; (void)n_in; (void)out_size; (void)ws_size;
    const float* pred = (const float*)d_in[0];
    const float* targ = (const float*)d_in[1];
    float* out     = (float*)d_out;
    float* partial = (float*)d_ws;          // nblk * 2 floats = 128 KB
    const int nblk = NCH * 256;             // 16384 tiles (64ch x 16x16 tiles)

    ssim_tile_kernel<<<nblk, 256, 0, stream>>>(pred, targ, partial);
    ssim_reduce_kernel<<<1, 256, 0, stream>>>(partial, nblk, out);
}